// SoftMultiFramePredictor_26474178413109
// MI455X (gfx1250) — compile-verified
//
#include <hip/hip_runtime.h>
#include <hip/hip_bf16.h>
#include <math.h>

// Problem constants (from the reference): B=1, N=2, C=3, H=128, W=256, K=31, T=15, DS=1
#define NN 2
#define CC 3
#define HH 128
#define WW 256
#define KK 31
#define TT 15
#define KK2 961            // K*K
#define LDS_ROW 33         // dwords per l-row: 32 positions + 1 pad => odd stride, conflict-free
#define BUF_FLOATS 1024    // one k-row buffer: 31*33 = 1023 -> 1024 dwords (4 KB)
#define WAVE_FLOATS 2048   // double buffer per wave (8 KB)

#define AS1 __attribute__((address_space(1)))
#define AS3 __attribute__((address_space(3)))

#if __has_builtin(__builtin_amdgcn_global_load_async_to_lds_b32)
#define USE_ASYNC 1
#else
#define USE_ASYNC 0
#endif

// s_wait_asynccnt requires a constant integer argument -> forward literals via macro.
#if __has_builtin(__builtin_amdgcn_s_wait_asynccnt)
#define WAIT_ASYNCCNT(n) __builtin_amdgcn_s_wait_asynccnt(n)
#else
#define WAIT_ASYNCCNT(n) asm volatile("s_wait_asynccnt %0" ::"i"(n) : "memory")
#endif

#if __has_builtin(__builtin_amdgcn_s_wait_dscnt)
#define WAIT_DSCNT0() __builtin_amdgcn_s_wait_dscnt(0)
#else
#define WAIT_DSCNT0() asm volatile("s_wait_dscnt 0x0" ::: "memory")
#endif

#if USE_ASYNC
// Stage one k-row (l = 0..30) of softmax weights for 32 positions into LDS.
// Per instruction p: lane L reads W[(pos0+p)*961 + k*31 + min(L,30)]  (contiguous 124B burst)
// and the DMA writes it into LDS row min(L,30), column p  (row stride 33 dwords -> all-distinct banks).
__device__ __forceinline__ void issue_batch(const float* __restrict__ weights,
                                            float* __restrict__ buf,
                                            int pos0, int k, int lane) {
  const int le = lane < (KK - 1) ? lane : (KK - 1);  // lane 31 duplicates lane 30 (no OOB, no LDS overflow)
  const float* g0 = weights + (size_t)pos0 * KK2 + (size_t)(k * KK + le);
  float* l0 = buf + le * LDS_ROW;
#pragma unroll
  for (int p = 0; p < 32; ++p) {
    __builtin_amdgcn_global_load_async_to_lds_b32(
        (AS1 int*)(g0 + (size_t)p * KK2),   // global source (AS1), const dropped
        (AS3 int*)(l0 + p),                 // LDS destination (AS3)
        /*imm offset*/ 0, /*cpol*/ 0);
  }
}
#endif

__global__ __launch_bounds__(256) void smfp_kernel(const float* __restrict__ frames,
                                                   const float* __restrict__ weights,
                                                   float* __restrict__ out) {
  const int lane = threadIdx.x & 31;
  const int wid = threadIdx.x >> 5;
  const int wgl = blockIdx.x * 8 + wid;   // global wave id, 2048 waves total
  const int pos0 = wgl << 5;              // first of 32 consecutive positions (same n,i row)
  const int n = pos0 >> 15;               // / (H*W)
  const int i = (pos0 >> 8) & (HH - 1);   // / W % H
  const int j = (pos0 & (WW - 1)) + lane; // this lane's output column

  __shared__ float lds[16384];            // 64 KB: 8 waves * 8 KB private double-buffer
  float* wbuf = &lds[wid * WAVE_FLOATS];

  float a0 = 0.f, a1 = 0.f, a2 = 0.f;     // 3 channel accumulators
  float px = 0.f, py = 0.f, an = 0.f, s = 0.f;

  const float* f0 = frames + (size_t)n * (CC * HH * WW);
  const int jm15 = j - TT;

#if USE_ASYNC
  issue_batch(weights, wbuf, pos0, 0, lane);
#endif

  for (int k = 0; k < KK; ++k) {
#if USE_ASYNC
    float* cur = wbuf + (k & 1) * BUF_FLOATS;
    if (k + 1 < KK) {
      WAIT_DSCNT0();  // our ds reads of the other buffer (iter k-1) are complete before overwrite
      issue_batch(weights, wbuf + ((k + 1) & 1) * BUF_FLOATS, pos0, k + 1, lane);
      WAIT_ASYNCCNT(32);  // async copies complete in order -> batch k fully landed
    } else {
      WAIT_ASYNCCNT(0);
    }
    // Deepen prefetch beyond the 2-batch ASYNCcnt window: pull row k+2 toward L2/WGP$.
    if (k + 2 < KK)
      __builtin_prefetch(weights + (size_t)(pos0 + lane) * KK2 + (size_t)((k + 2) * KK), 0, 1);
#else
    (void)wbuf;
#endif
    int row = i + k - TT;                       // edge ("replicate") padding
    row = row < 0 ? 0 : (row >= HH ? HH - 1 : row);
    const float* r0 = f0 + (size_t)row * WW;    // c = 0
    const float* r1 = r0 + HH * WW;             // c = 1
    const float* r2 = r1 + HH * WW;             // c = 2
    const float dk = (float)(k - TT);
    const float dk2 = dk * dk;
    float es = 0.f;

#pragma unroll
    for (int l = 0; l < KK; ++l) {
#if USE_ASYNC
      const float w = cur[l * LDS_ROW + lane];  // conflict-free, coalesced ds_load_b32
#else
      const float w = weights[(size_t)(pos0 + lane) * KK2 + (size_t)(k * KK + l)];
#endif
      const float e = __expf(w);                // unnormalized softmax (|w| << 1, exact up to fp)
      int col = jm15 + l;                       // edge padding on columns
      col = col < 0 ? 0 : (col >= WW ? WW - 1 : col);
      a0 = fmaf(e, r0[col], a0);
      a1 = fmaf(e, r1[col], a1);
      a2 = fmaf(e, r2[col], a2);
      const float dl = (float)(l - TT);
      px = fmaf(e, dl, px);                     // positions[...,0] = sum w * pos[l]  (after [::-1])
      an = fmaf(e, __builtin_sqrtf(dk2 + dl * dl), an);
      es += e;
    }
    py = fmaf(es, dk, py);                      // positions[...,1] = sum w * pos[k]
    s += es;                                    // softmax denominator
  }

  const float inv = 1.0f / s;
  const int hw = HH * WW;
  float* outO = out;                      // (1,N,C,H,W)  196608 floats
  float* outP = out + NN * CC * hw;       // (1,N,H,W,2)  131072 floats
  float* outE = outP + NN * hw * 2;       // (1,N,H,W)     65536 floats

  const size_t oij = (size_t)i * WW + j;
  outO[((size_t)(n * CC + 0) * HH) * WW + oij] = a0 * inv;
  outO[((size_t)(n * CC + 1) * HH) * WW + oij] = a1 * inv;
  outO[((size_t)(n * CC + 2) * HH) * WW + oij] = a2 * inv;
  const size_t pidx = (size_t)n * hw + oij;
  outP[pidx * 2 + 0] = px * inv;
  outP[pidx * 2 + 1] = py * inv;
  outE[pidx] = an * inv;
}

extern "C" void kernel_launch(void* const* d_in, const int* in_sizes, int n_in,
                              void* d_out, int out_size, void* d_ws, size_t ws_size,
                              hipStream_t stream) {
  (void)in_sizes; (void)n_in; (void)out_size; (void)d_ws; (void)ws_size;
  const float* frames = (const float*)d_in[0];   // (1,2,3,128,256) f32
  const float* weights = (const float*)d_in[1];  // (1,2,128,256,31,31) f32
  float* out = (float*)d_out;                    // 393216 f32 (out ++ positions ++ exp_norm)

  // 65536 output positions / 32 per wave = 2048 waves; 8 waves (256 threads) per block.
  smfp_kernel<<<dim3(256), dim3(256), 0, stream>>>(frames, weights, out);
}